// DisLoss_65180423684879
// MI455X (gfx1250) — compile-verified
//
#include <hip/hip_runtime.h>
#include <math.h>

typedef __attribute__((ext_vector_type(2))) float v2f;
typedef __attribute__((ext_vector_type(8))) float v8f;

#define NUM_CLS   4096
#define FEAT      256
#define BATCH     8192
#define INV_T     10.0f        // 1 / TEMPERATURE
#define COL_CHUNKS 8           // column partitions of the fused GEMM
#define TILES_PER_CHUNK (NUM_CLS / 16 / COL_CHUNKS)   // 32 col tiles per block

// ---------------------------------------------------------------------------
// Kernel 1: per-class sequential EMA chains (4096 independent chains).
// One workgroup per class, 256 threads = one feature dim each.
// wave32 ballot compaction keeps serial work proportional to #matches.
// ---------------------------------------------------------------------------
__global__ __launch_bounds__(256) void ema_kernel(const float* __restrict__ feat,
                                                  const int*   __restrict__ labels,
                                                  const float* __restrict__ pin,
                                                  float*       __restrict__ pout) {
    const int c    = blockIdx.x;
    const int tid  = threadIdx.x;
    const int wave = tid >> 5;
    const int lane = tid & 31;

    __shared__ unsigned int masks[8];
    __shared__ int   list[256];
    __shared__ int   cnt;
    __shared__ float wred[8];
    __shared__ float invn;

    float p = pin[c * FEAT + tid];

    for (int base = 0; base < BATCH; base += 256) {
        unsigned int m = __builtin_amdgcn_ballot_w32(labels[base + tid] == c);
        if (lane == 0) masks[wave] = m;
        __syncthreads();
        if (tid == 0) {                       // ordered compaction of matches
            int n = 0;
            #pragma unroll
            for (int w = 0; w < 8; ++w) {
                unsigned int mm = masks[w];
                while (mm) {
                    int b = __builtin_ctz(mm);
                    list[n++] = base + w * 32 + b;
                    mm &= mm - 1;
                }
            }
            cnt = n;
        }
        __syncthreads();
        const int n = cnt;

        for (int s = 0; s < n; ++s) {         // sequential (order-preserving) EMA
            float f   = feat[list[s] * FEAT + tid];
            float tmp = fmaf(p, 0.95f, f * 0.05f);
            float ss  = tmp * tmp;
            ss += __shfl_xor(ss, 1, 32);
            ss += __shfl_xor(ss, 2, 32);
            ss += __shfl_xor(ss, 4, 32);
            ss += __shfl_xor(ss, 8, 32);
            ss += __shfl_xor(ss, 16, 32);
            if (lane == 0) wred[wave] = ss;
            __syncthreads();
            if (tid == 0) {
                float t = 0.0f;
                #pragma unroll
                for (int w = 0; w < 8; ++w) t += wred[w];
                invn = 1.0f / fmaxf(sqrtf(t), 1e-12f);
            }
            __syncthreads();
            p = tmp * invn;
        }
        __syncthreads();                      // protect masks/list for next chunk
    }
    pout[c * FEAT + tid] = p;
}

// ---------------------------------------------------------------------------
// Kernel 2: fused P @ P^T * (1/T) -> exp -> mask diagonal -> per-row partial
// sums. Grid (32 row-blocks x 8 col-chunks) = 256 blocks, 8 waves each.
// Each wave owns 16 rows, sweeps its 32 col tiles 8 at a time (8x A reuse)
// with V_WMMA_F32_16X16X4_F32, and stores one partial per row (no atomics).
// Arguments to exp are bounded by |p_i . p_j| / T <= 10, so the fast
// hardware path (v_exp_f32) is exact enough: rel err ~1e-6 -> loss err ~1e-6.
// ---------------------------------------------------------------------------
__global__ __launch_bounds__(256) void gemm_rowsum_kernel(const float* __restrict__ P,
                                                          float*       __restrict__ part) {
    const int lane = threadIdx.x & 31;
    const int wave = threadIdx.x >> 5;
    const int half = lane >> 4;      // 0: K k,k+1 / rows M 0-7 ; 1: K k+2,k+3 / rows M 8-15
    const int l16  = lane & 15;
    const int rowBase = blockIdx.x * 128 + wave * 16;
    const int chunk   = blockIdx.y;
    const int ct0     = chunk * TILES_PER_CHUNK;

    const float* arow = P + (rowBase + l16) * FEAT + half * 2;

    float rs[8];
    #pragma unroll
    for (int r = 0; r < 8; ++r) rs[r] = 0.0f;

    for (int ct = ct0; ct < ct0 + TILES_PER_CHUNK; ct += 8) {
        const float* bbase = P + (ct * 16 + l16) * FEAT + half * 2;

        v8f acc[8];
        #pragma unroll
        for (int j = 0; j < 8; ++j) acc[j] = (v8f){};

        #pragma unroll 2
        for (int k = 0; k < FEAT; k += 4) {
            v2f a = *(const v2f*)(arow + k);
            #pragma unroll
            for (int j = 0; j < 8; ++j) {
                v2f b = *(const v2f*)(bbase + j * (16 * FEAT) + k);
                acc[j] = __builtin_amdgcn_wmma_f32_16x16x4_f32(false, a, false, b,
                                                               (short)0, acc[j], false, false);
            }
        }

        #pragma unroll
        for (int j = 0; j < 8; ++j) {
            const int gcol = (ct + j) * 16 + l16;
            #pragma unroll
            for (int r = 0; r < 8; ++r) {
                const int grow = rowBase + r + 8 * half;
                float e = __expf(acc[j][r] * INV_T);   // v_exp_f32 fast path
                rs[r] += (grow == gcol) ? 0.0f : e;
            }
        }
    }

    // reduce the 16 column-lanes of each half-wave; one partial store per row
    #pragma unroll
    for (int r = 0; r < 8; ++r) {
        float v = rs[r];
        v += __shfl_xor(v, 1, 32);
        v += __shfl_xor(v, 2, 32);
        v += __shfl_xor(v, 4, 32);
        v += __shfl_xor(v, 8, 32);
        if (l16 == 0) part[chunk * NUM_CLS + rowBase + r + 8 * half] = v;
    }
}

// ---------------------------------------------------------------------------
// Kernel 3: loss = mean_i log( (sum_k part[k][i]) / (C-1) )   (T/baseT == 1)
// ---------------------------------------------------------------------------
__global__ __launch_bounds__(256) void final_kernel(const float* __restrict__ part,
                                                    float*       __restrict__ out) {
    __shared__ float red[256];
    const int tid = threadIdx.x;
    float acc = 0.0f;
    for (int i = tid; i < NUM_CLS; i += 256) {
        float s = 0.0f;
        #pragma unroll
        for (int k = 0; k < COL_CHUNKS; ++k) s += part[k * NUM_CLS + i];
        acc += logf(s * (1.0f / (float)(NUM_CLS - 1)));
    }
    red[tid] = acc;
    __syncthreads();
    #pragma unroll
    for (int s = 128; s > 0; s >>= 1) {
        if (tid < s) red[tid] += red[tid + s];
        __syncthreads();
    }
    if (tid == 0) out[0] = red[0] * (1.0f / (float)NUM_CLS);
}

// ---------------------------------------------------------------------------
extern "C" void kernel_launch(void* const* d_in, const int* in_sizes, int n_in,
                              void* d_out, int out_size, void* d_ws, size_t ws_size,
                              hipStream_t stream) {
    const float* features   = (const float*)d_in[0];   // [8192, 256] f32
    const int*   labels     = (const int*)  d_in[1];   // [8192] i32 (JAX canonicalized)
    const float* prototypes = (const float*)d_in[2];   // [4096, 256] f32

    float* protos_ws = (float*)d_ws;                   // 4096*256 f32 = 4 MB
    float* part      = protos_ws + NUM_CLS * FEAT;     // [8][4096] f32 = 128 KB

    ema_kernel<<<NUM_CLS, 256, 0, stream>>>(features, labels, prototypes, protos_ws);
    gemm_rowsum_kernel<<<dim3(NUM_CLS / 128, COL_CHUNKS), 256, 0, stream>>>(protos_ws, part);
    final_kernel<<<1, 256, 0, stream>>>(part, (float*)d_out);
}